// FACT_48653389529750
// MI455X (gfx1250) — compile-verified
//
#include <hip/hip_runtime.h>
#include <hip/hip_bf16.h>
#include <math.h>

#define T_FRAMES 24000
#define HFEAT    909
#define COFF     512           // H - (NC1+NC2)
#define NC1      97
#define NC2      300
#define NCC      397
#define NACT     3806
#define SMSTRIDE 400

typedef float v2f __attribute__((ext_vector_type(2)));
typedef float v8f __attribute__((ext_vector_type(8)));

#if defined(__gfx1250__) && __has_builtin(__builtin_amdgcn_wmma_f32_16x16x4_f32)
#define HAVE_WMMA_F32X4 1
#else
#define HAVE_WMMA_F32X4 0
#endif

#if defined(__gfx1250__) && __has_builtin(__builtin_amdgcn_global_load_async_to_lds_b32) && __has_builtin(__builtin_amdgcn_s_wait_asynccnt)
#define HAVE_ASYNC_LDS 1
#else
#define HAVE_ASYNC_LDS 0
#endif

// ---------- wave32 cross-lane reductions ----------
__device__ __forceinline__ float waveMax(float v) {
    #pragma unroll
    for (int o = 16; o > 0; o >>= 1) v = fmaxf(v, __shfl_xor(v, o, 32));
    return v;
}
__device__ __forceinline__ float waveSum(float v) {
    #pragma unroll
    for (int o = 16; o > 0; o >>= 1) v += __shfl_xor(v, o, 32);
    return v;
}

// ---------- block-wide (256 threads = 8 waves) reductions ----------
__device__ __forceinline__ float blkMax256(float v, float* red8) {
    v = waveMax(v);
    int wave = threadIdx.x >> 5, lane = threadIdx.x & 31;
    if (lane == 0) red8[wave] = v;
    __syncthreads();
    float r = red8[0];
    #pragma unroll
    for (int w = 1; w < 8; ++w) r = fmaxf(r, red8[w]);
    __syncthreads();
    return r;
}
__device__ __forceinline__ float blkSum256(float v, float* red8) {
    v = waveSum(v);
    int wave = threadIdx.x >> 5, lane = threadIdx.x & 31;
    if (lane == 0) red8[wave] = v;
    __syncthreads();
    float r = red8[0];
    #pragma unroll
    for (int w = 1; w < 8; ++w) r += red8[w];
    __syncthreads();
    return r;
}

// ---------- kernel 0: pack (vid,nid) into one int ----------
__global__ void fact_pack_kernel(const int* __restrict__ vids,
                                 const int* __restrict__ nids,
                                 int* __restrict__ codes) {
    int a = blockIdx.x * blockDim.x + threadIdx.x;
    if (a < NACT) codes[a] = (vids[a] & 127) | (nids[a] << 7);
}

// ---------- kernel 1: per-frame LSE + action argmax ----------
__global__ void fact_frame_kernel(const float* __restrict__ ff,
                                  const int* __restrict__ codes,
                                  float* __restrict__ fLSE,
                                  int* __restrict__ pred) {
    int t   = blockIdx.x;
    int tid = threadIdx.x;
    __shared__ float fc[SMSTRIDE];
    __shared__ float red8[8];
    __shared__ float redw[8];
    __shared__ int   redwi[8];

    const float* row = ff + (size_t)t * HFEAT + COFF;
    for (int j = tid; j < NCC; j += 256) fc[j] = row[j];
    __syncthreads();

    // verb LSE over fc[0..96]
    float m = -3.402823466e+38f;
    for (int j = tid; j < NC1; j += 256) m = fmaxf(m, fc[j]);
    m = blkMax256(m, red8);
    float s = 0.f;
    for (int j = tid; j < NC1; j += 256) s += expf(fc[j] - m);
    s = blkSum256(s, red8);
    float vLSE = m + logf(s);

    // noun LSE over fc[97..396]
    m = -3.402823466e+38f;
    for (int j = tid; j < NC2; j += 256) m = fmaxf(m, fc[NC1 + j]);
    m = blkMax256(m, red8);
    s = 0.f;
    for (int j = tid; j < NC2; j += 256) s += expf(fc[NC1 + j] - m);
    s = blkSum256(s, red8);
    float nLSE = m + logf(s);

    // argmax over actions (LSE shift does not change argmax); first-index ties
    float best = -3.402823466e+38f;
    int   bi   = 0x7fffffff;
    for (int a = tid; a < NACT; a += 256) {
        int c = codes[a];
        float sc2 = fc[c & 127] + fc[NC1 + (c >> 7)];
        if (sc2 > best || (sc2 == best && a < bi)) { best = sc2; bi = a; }
    }
    #pragma unroll
    for (int o = 16; o > 0; o >>= 1) {
        float ov = __shfl_xor(best, o, 32);
        int   oi = __shfl_xor(bi, o, 32);
        if (ov > best || (ov == best && oi < bi)) { best = ov; bi = oi; }
    }
    int wave = tid >> 5, lane = tid & 31;
    if (lane == 0) { redw[wave] = best; redwi[wave] = bi; }
    __syncthreads();
    if (tid == 0) {
        float rb = redw[0]; int ri = redwi[0];
        #pragma unroll
        for (int w = 1; w < 8; ++w) {
            float ov = redw[w]; int oi = redwi[w];
            if (ov > rb || (ov == rb && oi < ri)) { rb = ov; ri = oi; }
        }
        pred[t] = ri;
        fLSE[t] = vLSE + nLSE;
    }
}

// ---------- kernel 2: scan -> seg_id, seg_start, nseg (single block, 1024 thr) ----------
__global__ void fact_scan_kernel(const int* __restrict__ pred,
                                 int* __restrict__ seg_id,
                                 int* __restrict__ seg_start,
                                 int* __restrict__ nseg_out) {
    const int C = 24;  // 1024 * 24 >= 24000
    __shared__ int sums[1024];
    int tid   = threadIdx.x;
    int begin = tid * C;
    int endi  = begin + C; if (endi > T_FRAMES) endi = T_FRAMES;

    int local = 0;
    for (int t = begin; t < endi; ++t) {
        int ch = (t == 0) ? 0 : (pred[t] != pred[t - 1]);
        local += ch;
    }
    sums[tid] = local;
    __syncthreads();
    for (int off = 1; off < 1024; off <<= 1) {
        int v   = sums[tid];
        int add = (tid >= off) ? sums[tid - off] : 0;
        __syncthreads();
        sums[tid] = v + add;
        __syncthreads();
    }
    int run = (tid > 0) ? sums[tid - 1] : 0;  // exclusive prefix
    for (int t = begin; t < endi; ++t) {
        int ch = (t == 0) ? 0 : (pred[t] != pred[t - 1]);
        run += ch;
        seg_id[t] = run;
        if (t == 0 || ch) seg_start[run] = t;
        if (t == T_FRAMES - 1) {
            nseg_out[0] = run + 1;
            seg_start[run + 1] = T_FRAMES;
        }
    }
}

// ---------- kernel 3: per-segment mean (WMMA f32 column-sum) + segment LSE ----------
__global__ void fact_seg_kernel(const float* __restrict__ ff,
                                const int* __restrict__ seg_start,
                                const int* __restrict__ nseg,
                                float* __restrict__ segmean,
                                float* __restrict__ sLSE) {
    int s = blockIdx.x;
    if (s >= *nseg) return;
    int start = seg_start[s];
    int end   = seg_start[s + 1];
    float invc = 1.0f / (float)(end - start);

    __shared__ float sc[SMSTRIDE];
    __shared__ float red8[8];
    int tid  = threadIdx.x;
    int lane = tid & 31;
    int wave = tid >> 5;

#if HAVE_WMMA_F32X4
    // Segment sum via V_WMMA_F32_16X16X4_F32 with A = ones(16x4):
    // C[m,n] = sum_k B[k,n] for every m, i.e. column sums of a
    // 4-frame x 16-feature tile. A is constant so the HW K-slot
    // permutation is irrelevant; N maps to lane&15 in C/D.
    for (int ft = wave; ft < 25; ft += 8) {          // 25 tiles of 16 features
        int feat  = ft * 16 + (lane & 15);
        int featc = (feat < NCC) ? feat : 0;         // safe address; lanes with
                                                     // feat>=397 feed unstored cols
        const float* base = ff + COFF + featc;
        int kbase = (lane >> 4) << 1;                // 0 or 2
        v8f acc = {};
        v2f a; a.x = 1.0f; a.y = 1.0f;
        for (int f0 = start; f0 < end; f0 += 4) {
            int fa = f0 + kbase, fb = fa + 1;
            int fac = (fa < end) ? fa : start;       // clamp address, mask value
            int fbc = (fb < end) ? fb : start;
            float va = base[(size_t)fac * HFEAT];
            float vb = base[(size_t)fbc * HFEAT];
            v2f b;
            b.x = (fa < end) ? va : 0.0f;
            b.y = (fb < end) ? vb : 0.0f;
            acc = __builtin_amdgcn_wmma_f32_16x16x4_f32(
                false, a, false, b, (short)0, acc, false, false);
        }
        if (lane < 16 && feat < NCC) sc[feat] = acc[0] * invc;
    }
#else
    for (int j = tid; j < NCC; j += 256) {
        float sum = 0.f;
        for (int f = start; f < end; ++f) sum += ff[(size_t)f * HFEAT + COFF + j];
        sc[j] = sum * invc;
    }
#endif
    __syncthreads();

    // segment verb/noun LSE over the means
    float m = -3.402823466e+38f;
    for (int j = tid; j < NC1; j += 256) m = fmaxf(m, sc[j]);
    m = blkMax256(m, red8);
    float su = 0.f;
    for (int j = tid; j < NC1; j += 256) su += expf(sc[j] - m);
    su = blkSum256(su, red8);
    float vLSE = m + logf(su);

    m = -3.402823466e+38f;
    for (int j = tid; j < NC2; j += 256) m = fmaxf(m, sc[NC1 + j]);
    m = blkMax256(m, red8);
    su = 0.f;
    for (int j = tid; j < NC2; j += 256) su += expf(sc[NC1 + j] - m);
    su = blkSum256(su, red8);
    float nLSE = m + logf(su);

    float* outrow = segmean + (size_t)s * SMSTRIDE;
    for (int j = tid; j < NCC; j += 256) outrow[j] = sc[j];
    if (tid == 0) sLSE[s] = vLSE + nLSE;
}

// ---------- kernel 4: output (store-bandwidth bound, 365 MB) ----------
__global__ void fact_out_kernel(const float* __restrict__ ff,
                                const int* __restrict__ codes,
                                const int* __restrict__ seg_id,
                                const float* __restrict__ fLSE,
                                const float* __restrict__ segmean,
                                const float* __restrict__ sLSE,
                                float* __restrict__ out) {
    int t   = blockIdx.x;
    int tid = threadIdx.x;
    int s   = seg_id[t];

    __shared__ float fcs[SMSTRIDE];
    __shared__ float scs[SMSTRIDE];
    __shared__ float comb[SMSTRIDE];

    const float* frow = ff + (size_t)t * HFEAT + COFF;
    const float* srow = segmean + (size_t)s * SMSTRIDE;

#if HAVE_ASYNC_LDS
    for (int j = tid; j < NCC; j += 256) {
        __builtin_amdgcn_global_load_async_to_lds_b32(
            (__attribute__((address_space(1))) int*)(frow + j),
            (__attribute__((address_space(3))) int*)(&fcs[j]), 0, 0);
        __builtin_amdgcn_global_load_async_to_lds_b32(
            (__attribute__((address_space(1))) int*)(srow + j),
            (__attribute__((address_space(3))) int*)(&scs[j]), 0, 0);
    }
    __builtin_amdgcn_s_wait_asynccnt(0);
    __syncthreads();
#else
    for (int j = tid; j < NCC; j += 256) { fcs[j] = frow[j]; scs[j] = srow[j]; }
    __syncthreads();
#endif

    float bias = 0.5f * (fLSE[t] + sLSE[s]);  // applied twice (verb + noun slots)
    for (int j = tid; j < NCC; j += 256) comb[j] = fcs[j] + scs[j] - bias;
    __syncthreads();

    size_t obase = (size_t)t * NACT;
    for (int a = tid; a < NACT; a += 256) {
        int c = codes[a];
        out[obase + a] = comb[c & 127] + comb[NC1 + (c >> 7)];
    }
}

extern "C" void kernel_launch(void* const* d_in, const int* in_sizes, int n_in,
                              void* d_out, int out_size, void* d_ws, size_t ws_size,
                              hipStream_t stream) {
    const float* ff   = (const float*)d_in[0];
    const int*   vids = (const int*)d_in[1];
    const int*   nids = (const int*)d_in[2];
    float*       out  = (float*)d_out;

    // bump allocator on workspace (256B aligned)
    char*  ws  = (char*)d_ws;
    size_t off = 0;
    auto alloc = [&](size_t bytes) -> void* {
        void* p = ws + off;
        off += bytes;
        off = (off + 255) & ~(size_t)255;
        return p;
    };

    float* fLSE      = (float*)alloc(sizeof(float) * T_FRAMES);
    int*   pred      = (int*)  alloc(sizeof(int)   * T_FRAMES);
    int*   seg_id    = (int*)  alloc(sizeof(int)   * T_FRAMES);
    int*   seg_start = (int*)  alloc(sizeof(int)   * (T_FRAMES + 1));
    int*   nseg      = (int*)  alloc(sizeof(int));
    int*   codes     = (int*)  alloc(sizeof(int)   * NACT);
    float* sLSE      = (float*)alloc(sizeof(float) * T_FRAMES);
    float* segmean   = (float*)alloc(sizeof(float) * (size_t)T_FRAMES * SMSTRIDE);
    (void)ws_size; (void)in_sizes; (void)n_in; (void)out_size;

    fact_pack_kernel<<<(NACT + 255) / 256, 256, 0, stream>>>(vids, nids, codes);
    fact_frame_kernel<<<T_FRAMES, 256, 0, stream>>>(ff, codes, fLSE, pred);
    fact_scan_kernel<<<1, 1024, 0, stream>>>(pred, seg_id, seg_start, nseg);
    fact_seg_kernel<<<T_FRAMES, 256, 0, stream>>>(ff, seg_start, nseg, segmean, sLSE);
    fact_out_kernel<<<T_FRAMES, 256, 0, stream>>>(ff, codes, seg_id, fLSE, segmean, sLSE, out);
}